// SelfSuperVisedLearningLoss_5523327943235
// MI455X (gfx1250) — compile-verified
//
#include <hip/hip_runtime.h>
#include <math.h>

#define DT     0.2f
#define SAFE_C 1.5f
#define PARK_C 5.0f
#define RAD_C  1.5f
#define DEFV_C 2.5f
#define B0     1.0f
#define B1     0.8f

typedef __attribute__((ext_vector_type(2))) float v2f;
typedef __attribute__((ext_vector_type(8))) float v8f;
typedef __attribute__((ext_vector_type(4))) int   v4i;

// ---------------------------------------------------------------------------
// Wave-level f32 sum through the matrix pipe (V_WMMA_F32_16X16X4_F32):
// A (16x4 f32, 2 VGPRs/lane) holds each lane's value in k-slot .x, zeros in .y;
// B = all ones -> D[m][n] = sum_k A[m][k] = v[lane m] + v[lane m+16].
// Each lane sums its 8 accumulator VGPRs (rows 0-7 or 8-15); shfl_xor(16)
// combines halves -> full 32-lane sum in every lane. Full f32 precision.
// EXEC must be all ones at the call site.
// ---------------------------------------------------------------------------
__device__ __forceinline__ float wave_sum_wmma(float v) {
  v2f a; a[0] = v;    a[1] = 0.0f;
  v2f b; b[0] = 1.0f; b[1] = 1.0f;
  v8f c = {};
  c = __builtin_amdgcn_wmma_f32_16x16x4_f32(false, a, false, b, (short)0, c,
                                            false, false);
  float s = ((c[0] + c[1]) + (c[2] + c[3])) + ((c[4] + c[5]) + (c[6] + c[7]));
  s += __shfl_xor(s, 16, 32);
  return s;
}

__device__ __forceinline__ void block_store_sum(float wsum, float* dst) {
  __shared__ float sh[32];
  const int lane = threadIdx.x & 31;
  const int wid  = threadIdx.x >> 5;
  if (lane == 0) sh[wid] = wsum;
  __syncthreads();
  if (threadIdx.x == 0) {
    const int nw = (blockDim.x + 31) >> 5;
    float s = 0.0f;
    for (int w = 0; w < nw; ++w) s += sh[w];
    *dst = s;
  }
}

// Non-returning relaxed agent-scope f32 add -> GLOBAL_ATOMIC_ADD_F32 (no CAS).
__device__ __forceinline__ void atomic_add_f32(float* p, float v) {
  (void)__hip_atomic_fetch_add(p, v, __ATOMIC_RELAXED, __HIP_MEMORY_SCOPE_AGENT);
}

// ---------------------------------------------------------------------------
// Kernel 1: per-node prep. Packs everything the edge pass gathers:
//   posrec[i] = {pos_pred.x, pos_pred.y, pos_ref.x, pos_ref.y}
//   aux[i]    = {speed, col7}
// and zeroes the per-vehicle collision-diff accumulator.
// psi_t (tan + mod 2pi) is dead code downstream and is skipped.
// ---------------------------------------------------------------------------
__global__ void k_prep(const float* __restrict__ yp, const float* __restrict__ yr,
                       const float* __restrict__ X, float4* __restrict__ posrec,
                       float2* __restrict__ aux, float* __restrict__ acc,
                       int N, int NV) {
  const int i = blockIdx.x * blockDim.x + threadIdx.x;
  if (i >= N) return;
  const float* row = X + (size_t)i * 9;
  const float c0 = row[0], c1 = row[1], c3 = row[3], c7 = row[7];
  if (i < NV) {
    const float c2  = row[2];
    const float yr0 = yr[2 * i], yr1 = yr[2 * i + 1];
    const float yp0 = yp[2 * i];
    const float xt = c0 + c3 * cosf(c2) * DT;
    const float yt = c1 + c3 * sinf(c2) * DT;
    const float vt = 0.99f * c3 + yr0 * DT;
    const float sp = vt + (yp0 - yr0) * DT;            // speed after pedal swap
    const float ct = cosf(yr1), st = sinf(yr1);
    posrec[i] = make_float4(xt + ct * sp * DT, yt + st * sp * DT, xt, yt);
    aux[i]    = make_float2(sp, c7);
    acc[i]    = 0.0f;
  } else {
    posrec[i] = make_float4(c0, c1, c0, c1);
    aux[i]    = make_float2(c3, c7);
  }
}

// ---------------------------------------------------------------------------
// Kernel 2: edge pass. segment_sum output is sliced [:nv], so edges with
// e1 >= NV are dead. Both col_loss calls share `safe`/`base`, and only the
// difference (pred - ref) survives -> ONE f32 atomic per live edge.
// Edge index stream (128 MB, read-once) uses non-temporal loads so the hot
// 16 MB node tables stay resident in the near caches.
// ---------------------------------------------------------------------------
__device__ __forceinline__ void edge_one(int e0, int e1,
                                         const float4* __restrict__ posrec,
                                         const float2* __restrict__ aux,
                                         float* __restrict__ acc, int NV) {
  if (e1 >= NV) return;                         // segment discarded by [:nv]
  const float4 p0 = posrec[e0];
  const float4 p1 = posrec[e1];
  const float2 a0 = aux[e0];
  const float2 a1 = aux[e1];
  const bool veh  = (e0 < NV);
  const float safe = fabsf(a1.x) + (veh ? fabsf(a0.x) : 0.0f) + SAFE_C;
  const float base = safe + a0.y + (veh ? 2.0f : 1.0f) * RAD_C;

  float dx = p0.x - p1.x, dy = p0.y - p1.y;
  const float dp = sqrtf(dx * dx + dy * dy);
  const float lp = fmaxf(base - dp, 0.0f);
  const float cp = lp * lp + lp;

  dx = p0.z - p1.z; dy = p0.w - p1.w;
  const float dr = sqrtf(dx * dx + dy * dy);
  const float lr = fmaxf(base - dr, 0.0f);
  const float cr = lr * lr + lr;

  const float diff = cp - cr;
  if (diff != 0.0f) atomic_add_f32(&acc[e1], diff);
}

__global__ void k_edges4(const int* __restrict__ e0p, const int* __restrict__ e1p,
                         const float4* __restrict__ posrec,
                         const float2* __restrict__ aux,
                         float* __restrict__ acc, int G, int NV) {
  const int t = blockIdx.x * blockDim.x + threadIdx.x;
  if (t >= G) return;
  const v4i a = __builtin_nontemporal_load(((const v4i*)e0p) + t);
  const v4i b = __builtin_nontemporal_load(((const v4i*)e1p) + t);
  edge_one(a[0], b[0], posrec, aux, acc, NV);
  edge_one(a[1], b[1], posrec, aux, acc, NV);
  edge_one(a[2], b[2], posrec, aux, acc, NV);
  edge_one(a[3], b[3], posrec, aux, acc, NV);
}

__global__ void k_edges1(const int* __restrict__ e0p, const int* __restrict__ e1p,
                         const float4* __restrict__ posrec,
                         const float2* __restrict__ aux,
                         float* __restrict__ acc, int E, int NV) {
  const int t = blockIdx.x * blockDim.x + threadIdx.x;
  if (t >= E) return;
  edge_one(e0p[t], e1p[t], posrec, aux, acc, NV);
}

// ---------------------------------------------------------------------------
// Kernel 3: per-vehicle loss + masks, block-reduced via WMMA to one partial
// per block. Each term is pre-scaled by 1/NV so the final sum is the mean.
// ---------------------------------------------------------------------------
__global__ void k_vehicle(const float* __restrict__ yp, const float* __restrict__ yr,
                          const float* __restrict__ X, const float* __restrict__ be,
                          const float4* __restrict__ posrec,
                          const float* __restrict__ acc,
                          float* __restrict__ partials, int NV) {
  const int i = blockIdx.x * blockDim.x + threadIdx.x;
  float t = 0.0f;
  if (i < NV) {
    const float* row = X + (size_t)i * 9;
    const float c3 = row[3], c4 = row[4], c5 = row[5];
    const float yr0 = yr[2 * i], yr1 = yr[2 * i + 1];
    const float yp0 = yp[2 * i], yp1 = yp[2 * i + 1];
    const float4 p = posrec[i];

    const float d = (sqrtf(p.x * p.x + p.y * p.y) -
                     sqrtf(p.z * p.z + p.w * p.w)) + acc[i];
    const float l22 = (d >= 0.0f) ? (d * d + d) : d;

    const float q0 = (yr0 - yp0) / B0;
    const float l21 = q0 * q0;

    const float vt  = 0.99f * c3 + yr0 * DT;
    const float ddx = p.z - c4, ddy = p.w - c5;
    const bool m1 = (sqrtf(ddx * ddx + ddy * ddy) - (PARK_C + fabsf(c3))) > 0.0f;
    const bool m2 = fabsf(vt) >= (DEFV_C - 1e-5f);
    const float sg = (vt > 0.0f) ? 1.0f : ((vt < 0.0f) ? -1.0f : 0.0f);
    const bool m3 = (sg * (yr0 - yp0)) <= 1e-5f;
    const bool mask = m1 && m2 && m3;

    const float l2 = mask ? l22 : l21;
    const float q1 = (yr1 - yp1) / B1;
    const float l1 = q1 * q1;
    const float b0 = be[2 * i] / B0, b1 = be[2 * i + 1] / B1;
    const float l3 = b0 * b0 + b1 * b1;

    t = (l1 + l2 + 0.1f * l3) * (1.0f / (float)NV);
  }
  // EXEC all-ones here: every thread reaches this point.
  const float wsum = wave_sum_wmma(t);
  block_store_sum(wsum, &partials[blockIdx.x]);
}

// Kernel 4: single-block final reduction of per-block partials.
__global__ void k_final(const float* __restrict__ partials, int nb,
                        float* __restrict__ out) {
  float t = 0.0f;
  for (int j = threadIdx.x; j < nb; j += blockDim.x) t += partials[j];
  const float wsum = wave_sum_wmma(t);
  block_store_sum(wsum, out);
}

// ---------------------------------------------------------------------------
extern "C" void kernel_launch(void* const* d_in, const int* in_sizes, int n_in,
                              void* d_out, int out_size, void* d_ws, size_t ws_size,
                              hipStream_t stream) {
  const float* y_pred = (const float*)d_in[0];
  const float* y_ref  = (const float*)d_in[1];
  const float* X_cur  = (const float*)d_in[2];
  const int*   edges  = (const int*)d_in[3];
  const float* be_st  = (const float*)d_in[4];
  float*       out    = (float*)d_out;

  const int NV = in_sizes[0] / 2;
  const int N  = in_sizes[2] / 9;
  const int E  = in_sizes[3] / 2;

  char* ws = (char*)d_ws;
  float4* posrec = (float4*)ws;                 ws += (size_t)N * sizeof(float4);
  float2* aux    = (float2*)ws;                 ws += (size_t)N * sizeof(float2);
  float*  acc    = (float*)ws;                  ws += (size_t)NV * sizeof(float);
  float*  parts  = (float*)ws;

  const int TB = 256;
  const int nb_prep = (N + TB - 1) / TB;
  k_prep<<<nb_prep, TB, 0, stream>>>(y_pred, y_ref, X_cur, posrec, aux, acc, N, NV);

  const int* e0p = edges;
  const int* e1p = edges + E;
  if ((E & 3) == 0) {
    const int G = E >> 2;
    k_edges4<<<(G + TB - 1) / TB, TB, 0, stream>>>(e0p, e1p, posrec, aux, acc, G, NV);
  } else {
    k_edges1<<<(E + TB - 1) / TB, TB, 0, stream>>>(e0p, e1p, posrec, aux, acc, E, NV);
  }

  const int nb_veh = (NV + TB - 1) / TB;
  k_vehicle<<<nb_veh, TB, 0, stream>>>(y_pred, y_ref, X_cur, be_st, posrec, acc,
                                       parts, NV);
  k_final<<<1, TB, 0, stream>>>(parts, nb_veh, out);
}